// TotalLoss_47347719471590
// MI455X (gfx1250) — compile-verified
//
#include <hip/hip_runtime.h>
#include <hip/hip_bf16.h>

// Problem constants (match reference)
#define BB 2
#define DD 64
#define HH 256
#define WW 256
#define HW 65536           // H*W
#define BD 128             // B*D
#define CH_STRIDE 4194304  // D*H*W
#define EPSF 1e-6f
#define ALPHAF 5.0f
#define PI_F 3.14159265358979323846f

// ---------------- CDNA5 async-LDS path (guarded, with safe fallback) -------
#ifndef HAVE_ASYNC_LDS
#if defined(__has_builtin)
#if __has_builtin(__builtin_amdgcn_global_load_async_to_lds_b128)
#define HAVE_ASYNC_LDS 1
#endif
#if __has_builtin(__builtin_amdgcn_s_wait_asynccnt)
#define HAVE_WAIT_ASYNC 1
#endif
#endif
#endif
#ifndef HAVE_ASYNC_LDS
#define HAVE_ASYNC_LDS 0
#endif
#ifndef HAVE_WAIT_ASYNC
#define HAVE_WAIT_ASYNC 0
#endif

// generic-vector int4, matching the builtin's parameter type
typedef int v4i_cdna5 __attribute__((__vector_size__(16)));

#if HAVE_ASYNC_LDS
#define ASYNC_CP16(gsrc, ldst)                                               \
  __builtin_amdgcn_global_load_async_to_lds_b128(                            \
      (__attribute__((address_space(1))) v4i_cdna5*)(gsrc),                  \
      (__attribute__((address_space(3))) v4i_cdna5*)(ldst), 0, 0)
#if HAVE_WAIT_ASYNC
#define WAIT_ASYNC_LE2() do { __builtin_amdgcn_s_wait_asynccnt(2); __asm__ volatile("" ::: "memory"); } while (0)
#define WAIT_ASYNC_LE0() do { __builtin_amdgcn_s_wait_asynccnt(0); __asm__ volatile("" ::: "memory"); } while (0)
#else
#define WAIT_ASYNC_LE2() __asm__ volatile("s_wait_asynccnt 2" ::: "memory")
#define WAIT_ASYNC_LE0() __asm__ volatile("s_wait_asynccnt 0" ::: "memory")
#endif
#else
#define ASYNC_CP16(gsrc, ldst) (*(float4*)(ldst) = *(const float4*)(gsrc))
#define WAIT_ASYNC_LE2() __asm__ volatile("" ::: "memory")
#define WAIT_ASYNC_LE0() __asm__ volatile("" ::: "memory")
#endif

// ---------------------------------------------------------------------------
// Kernel 3 (defined FIRST so the disasm snippet shows the async pipeline):
// main per-(b,d) reduction. 1 block = one 8192-pixel chunk of one (b,d)
// slice, both channels. Async double-buffered LDS pipeline, 8 stages.
// Writes 16 partial sums per block to Cp[block][16].
// acc layout: 0-2 art0*(p,py,px); 3-5 art1*(p,py,px); 6 p0sum; 7 p1sum;
//             8-11 k=0 (lu_out*p0, ow_out*p1, lu_in*(1-p0), ow_in*(1-p1));
//             12-15 same for k=1.
// ---------------------------------------------------------------------------
__global__ __launch_bounds__(256) void k_main(const float* __restrict__ L,
                                              const float* __restrict__ T,
                                              const float* __restrict__ rad,
                                              float* __restrict__ Cp) {
  __shared__ float4 buf[2][2][256];   // [stage parity][channel][thread]
  __shared__ float wsum[8][16];
  const int tid = threadIdx.x;
  const int bd = blockIdx.x >> 3;
  const int chunk = blockIdx.x & 7;
  const int b = bd >> 6, d = bd & (DD - 1);

  const float ty0 = T[bd],        ty1 = T[BD + bd];
  const float tx0 = T[256 + bd],  tx1 = T[256 + BD + bd];
  const float r00 = rad[0], r01 = rad[1], r10 = rad[2], r11 = rad[3];

  const int base0 = (b * 2 * DD + d) * HW + chunk * 8192;  // channel 0
  const int base1 = base0 + CH_STRIDE;                     // channel 1

  float acc[16];
#pragma unroll
  for (int i = 0; i < 16; ++i) acc[i] = 0.0f;

#define ISSUE_STAGE(s)                                        \
  do {                                                        \
    const float* g0 = L + base0 + (s) * 1024 + tid * 4;       \
    const float* g1 = L + base1 + (s) * 1024 + tid * 4;       \
    ASYNC_CP16(g0, &buf[(s) & 1][0][tid]);                    \
    ASYNC_CP16(g1, &buf[(s) & 1][1][tid]);                    \
  } while (0)

  ISSUE_STAGE(0);
  for (int s = 0; s < 8; ++s) {
    if (s < 7) {
      ISSUE_STAGE(s + 1);
      WAIT_ASYNC_LE2();   // stage s's 2 copies complete (in-order)
    } else {
      WAIT_ASYNC_LE0();
    }
    // each lane consumes exactly the data it copied -> no cross-wave barrier
    float4 v0 = buf[s & 1][0][tid];
    float4 v1 = buf[s & 1][1][tid];
    const int pixbase = chunk * 8192 + s * 1024 + tid * 4;
    float l0a[4] = {v0.x, v0.y, v0.z, v0.w};
    float l1a[4] = {v1.x, v1.y, v1.z, v1.w};
#pragma unroll
    for (int j = 0; j < 4; ++j) {
      const int pix = pixbase + j;
      const float fh = (float)(pix >> 8);
      const float fw = (float)(pix & (WW - 1));
      const float p0 = 1.0f / (1.0f + expf(-l0a[j]));
      const float p1 = 1.0f / (1.0f + expf(-l1a[j]));
      // distances to both targets
      const float dy0 = fh - ty0 + EPSF, dx0 = fw - tx0 + EPSF;
      const float dy1 = fh - ty1 + EPSF, dx1 = fw - tx1 + EPSF;
      const float dd0 = sqrtf(dy0 * dy0 + dx0 * dx0);
      const float dd1 = sqrtf(dy1 * dy1 + dx1 * dx1);
      // swapped artery masks: arts[k] = (d2s[1-k] > radiuses[1-k,1])
      const float art0 = (dd1 > r11) ? 1.0f : 0.0f;
      const float art1 = (dd0 > r01) ? 1.0f : 0.0f;
      const float ps = p0 + p1;
      acc[0] += art0 * ps;  acc[1] += art0 * fh * ps;  acc[2] += art0 * fw * ps;
      acc[3] += art1 * ps;  acc[4] += art1 * fh * ps;  acc[5] += art1 * fw * ps;
      acc[6] += p0;         acc[7] += p1;
      // k=0 constraint terms (d2 = dd0, art = art0, radii r00/r01)
      const float e_lu0 = expf(1.0f - dd0 / r00);
      const float e_ow0 = expf(1.0f - dd0 / r01);
      acc[8]  += fmaxf(1.0f - e_lu0, 0.0f) * art0 * p0;
      acc[9]  += fmaxf(1.0f - e_ow0, 0.0f) * art0 * p1;
      acc[10] += fmaxf(e_lu0 - 1.0f, 0.0f) * (1.0f - p0);
      acc[11] += fmaxf(e_ow0 - 1.0f, 0.0f) * (1.0f - p1);
      // k=1 constraint terms (d2 = dd1, art = art1, radii r10/r11)
      const float e_lu1 = expf(1.0f - dd1 / r10);
      const float e_ow1 = expf(1.0f - dd1 / r11);
      acc[12] += fmaxf(1.0f - e_lu1, 0.0f) * art1 * p0;
      acc[13] += fmaxf(1.0f - e_ow1, 0.0f) * art1 * p1;
      acc[14] += fmaxf(e_lu1 - 1.0f, 0.0f) * (1.0f - p0);
      acc[15] += fmaxf(e_ow1 - 1.0f, 0.0f) * (1.0f - p1);
    }
  }
#undef ISSUE_STAGE

  // wave32 reduction, then across the 8 waves via LDS (deterministic)
#pragma unroll
  for (int i = 0; i < 16; ++i) {
    float v = acc[i];
    for (int off = 16; off > 0; off >>= 1) v += __shfl_down(v, off, 32);
    if ((tid & 31) == 0) wsum[tid >> 5][i] = v;
  }
  __syncthreads();
  if (tid < 16) {
    float s = 0.0f;
    for (int w = 0; w < 8; ++w) s += wsum[w][tid];
    Cp[blockIdx.x * 16 + tid] = s;
  }
}

// ---------------------------------------------------------------------------
// Kernel 1: extract centerline targets via atomics into A[ cnt|ys|xs ][2][BD]
// ---------------------------------------------------------------------------
__global__ __launch_bounds__(256) void k_extract(const int4* __restrict__ cl4,
                                                 float* __restrict__ A) {
  int t = blockIdx.x * 256 + threadIdx.x;   // 0 .. (B*D*H*W/4 - 1)
  int4 v = cl4[t];
  int base = t * 4;
  int bd = base >> 16;       // (b*D + d), same for all 4 elems
  int d = bd & (DD - 1);
  if (d == 0) return;        // reference zeroes d==0 slice before extraction
  int vv[4] = {v.x, v.y, v.z, v.w};
#pragma unroll
  for (int j = 0; j < 4; ++j) {
    int val = vv[j];
    if (val == 1 || val == 2) {
      int pix = (base + j) & (HW - 1);
      int k = val - 1;
      atomicAdd(&A[k * BD + bd], 1.0f);
      atomicAdd(&A[2 * BD + k * BD + bd], (float)(pix >> 8));
      atomicAdd(&A[4 * BD + k * BD + bd], (float)(pix & (WW - 1)));
    }
  }
}

// ---------------------------------------------------------------------------
// Kernel 2: finalize targets -> T = [ ty[2][BD] | tx[2][BD] | valid[2][BD] ]
// ---------------------------------------------------------------------------
__global__ __launch_bounds__(256) void k_ftgt(const float* __restrict__ A,
                                              float* __restrict__ T) {
  int t = threadIdx.x;  // 0..255 covers [k][bd]
  float cnt = A[t];
  bool one = (cnt == 1.0f);
  T[t]           = one ? A[256 + t] : -1.0f;  // ty
  T[256 + t]     = one ? A[512 + t] : -1.0f;  // tx
  T[512 + t]     = one ? 1.0f : 0.0f;         // valid
}

// ---------------------------------------------------------------------------
// Kernel 4: finalize cent + cnst. One block, 128 threads (one per (b,d)).
// E[0] = 0.02*loss_cent + loss_cnst
// ---------------------------------------------------------------------------
__global__ __launch_bounds__(128) void k_fmain(const float* __restrict__ Cp,
                                               const float* __restrict__ T,
                                               const float* __restrict__ rad,
                                               float* __restrict__ E) {
  const int bd = threadIdx.x;  // 0..127
  float S[16];
#pragma unroll
  for (int i = 0; i < 16; ++i) {
    float s = 0.0f;
    for (int c = 0; c < 8; ++c) s += Cp[(bd * 8 + c) * 16 + i];
    S[i] = s;
  }
  const float ty0 = T[bd], ty1 = T[BD + bd];
  const float tx0 = T[256 + bd], tx1 = T[256 + BD + bd];
  const float v0 = T[512 + bd], v1 = T[512 + BD + bd];
  const float r00 = rad[0], r01 = rad[1], r10 = rad[2], r11 = rad[3];

  // loss_cent contributions
  float den0 = S[0] + EPSF;
  float dy = S[1] / den0 - ty0 + EPSF;
  float dx = S[2] / den0 - tx0 + EPSF;
  double cent = (double)(sqrtf(dy * dy + dx * dx) * v0);
  float den1 = S[3] + EPSF;
  dy = S[4] / den1 - ty1 + EPSF;
  dx = S[5] / den1 - tx1 + EPSF;
  cent += (double)(sqrtf(dy * dy + dx * dx) * v1);

  // loss_cnst contributions (p_lu = channel0 sums, p_ow = channel1 sums)
  const float sp0 = S[6] + EPSF, sp1 = S[7] + EPSF;
  double cnst = (double)v0 * ((double)(S[8] / ALPHAF) / sp0 +
                              (double)(S[9] / ALPHAF) / sp1 +
                              (double)S[10] / (r00 * r00 * PI_F) +
                              (double)S[11] / (r01 * r01 * PI_F));
  cnst += (double)v1 * ((double)(S[12] / ALPHAF) / sp0 +
                        (double)(S[13] / ALPHAF) / sp1 +
                        (double)S[14] / (r10 * r10 * PI_F) +
                        (double)S[15] / (r11 * r11 * PI_F));

  __shared__ double rc[128], rn[128];
  rc[bd] = cent;
  rn[bd] = cnst;
  __syncthreads();
  for (int off = 64; off > 0; off >>= 1) {
    if (bd < off) { rc[bd] += rc[bd + off]; rn[bd] += rn[bd + off]; }
    __syncthreads();
  }
  if (bd == 0) {
    double loss_cent = rc[0] / (double)BD;  // l2.mean() summed over k
    double loss_cnst = rn[0] / (double)BD;  // (rc_out+rc_in).mean() over k
    E[0] = (float)(0.02 * loss_cent + 1.0 * loss_cnst);
  }
}

// ---------------------------------------------------------------------------
// Kernel 5: CSIS partial sums. 2048 blocks x 256 threads, grid-stride over
// B * 62 * H * W elements; per-block partial -> Dp[block].
// ---------------------------------------------------------------------------
__global__ __launch_bounds__(256) void k_csis(const float* __restrict__ L,
                                              float* __restrict__ Dp) {
  const float invT = 1.0f / 0.7f;
  const int ELEMS = BB * (DD - 2) * HW;  // 8,126,464
  const int STEP = 2048 * 256;
  float accv = 0.0f;
  for (int q = blockIdx.x * 256 + threadIdx.x; q < ELEMS; q += STEP) {
    const int b = (q >= (DD - 2) * HW) ? 1 : 0;
    const int rem = q - b * (DD - 2) * HW;
    const int ds = rem >> 16;          // shifted slice 0..61
    const int pix = rem & (HW - 1);
    const int base = (b * 2 * DD + ds) * HW + pix;
    if (q + STEP < ELEMS) {  // CDNA5 global_prefetch_b8 for next iteration
      __builtin_prefetch(L + base + STEP, 0, 0);
      __builtin_prefetch(L + base + CH_STRIDE + STEP, 0, 0);
    }
    const float l0s = L[base];
    const float l1s = L[base + CH_STRIDE];
    const float l0o = L[base + HW];               // orig = slice ds+1
    const float l1o = L[base + CH_STRIDE + HW];
    const float as_ = (l0s - l1s) * invT;
    const float ao_ = (l0o - l1o) * invT;
    const float ps0 = 1.0f / (1.0f + expf(-as_));
    const float ps1 = 1.0f / (1.0f + expf(as_));
    const float po0 = 1.0f / (1.0f + expf(-ao_));
    const float po1 = 1.0f / (1.0f + expf(ao_));
    const float lp0 = fmaxf(logf(po0), -100.0f);
    const float lm0 = fmaxf(logf(1.0f - po0), -100.0f);
    const float lp1 = fmaxf(logf(po1), -100.0f);
    const float lm1 = fmaxf(logf(1.0f - po1), -100.0f);
    accv += ps0 * lp0 + (1.0f - ps0) * lm0 + ps1 * lp1 + (1.0f - ps1) * lm1;
  }
  for (int off = 16; off > 0; off >>= 1) accv += __shfl_down(accv, off, 32);
  __shared__ float wpart[8];
  if ((threadIdx.x & 31) == 0) wpart[threadIdx.x >> 5] = accv;
  __syncthreads();
  if (threadIdx.x == 0) {
    float s = 0.0f;
    for (int w = 0; w < 8; ++w) s += wpart[w];
    Dp[blockIdx.x] = s;
  }
}

// ---------------------------------------------------------------------------
// Kernel 6: combine csis partials with E[0] -> final scalar
// ---------------------------------------------------------------------------
__global__ __launch_bounds__(256) void k_combine(const float* __restrict__ Dp,
                                                 const float* __restrict__ E,
                                                 float* __restrict__ out) {
  const int t = threadIdx.x;
  double s = 0.0;
  for (int i = t; i < 2048; i += 256) s += (double)Dp[i];
  __shared__ double red[256];
  red[t] = s;
  __syncthreads();
  for (int off = 128; off > 0; off >>= 1) {
    if (t < off) red[t] += red[t + off];
    __syncthreads();
  }
  if (t == 0) {
    const double count = (double)(BB * 2 * (DD - 2)) * (double)HW;  // 16,252,928
    const double loss_csis = -red[0] / count;
    out[0] = (float)((double)E[0] + 0.001 * loss_csis);
  }
}

// ---------------------------------------------------------------------------
extern "C" void kernel_launch(void* const* d_in, const int* in_sizes, int n_in,
                              void* d_out, int out_size, void* d_ws, size_t ws_size,
                              hipStream_t stream) {
  const float* logits = (const float*)d_in[0];
  const int*   cl     = (const int*)d_in[1];
  const float* rad    = (const float*)d_in[2];
  float* ws = (float*)d_ws;

  // workspace layout (floats)
  float* A  = ws;            // 768: atomic cnt/ys/xs [2][BD] x3
  float* T  = ws + 768;      // 768: ty/tx/valid [2][BD] x3
  float* Cp = ws + 1536;     // 16384: main partials [BD*8][16]
  float* Dp = ws + 17920;    // 2048: csis partials
  float* E  = ws + 19968;    // 2: cent+cnst combined

  (void)hipMemsetAsync(A, 0, 768 * sizeof(float), stream);

  // 1) target extraction (int4-vectorized scan of the mask)
  k_extract<<<(BB * DD * HW / 4) / 256, 256, 0, stream>>>((const int4*)cl, A);
  // 2) finalize targets
  k_ftgt<<<1, 256, 0, stream>>>(A, T);
  // 3) main fused reduction: 128 (b,d) slices x 8 chunks
  k_main<<<BD * 8, 256, 0, stream>>>(logits, T, rad, Cp);
  // 4) cent + cnst finalize
  k_fmain<<<1, 128, 0, stream>>>(Cp, T, rad, E);
  // 5) csis partials
  k_csis<<<2048, 256, 0, stream>>>(logits, Dp);
  // 6) final combine -> scalar
  k_combine<<<1, 256, 0, stream>>>(Dp, E, (float*)d_out);
  (void)in_sizes; (void)n_in; (void)out_size; (void)ws_size;
}